// SwapPrecision_3135326126282
// MI455X (gfx1250) — compile-verified
//
#include <hip/hip_runtime.h>

#define N 4096
#define D 128
#define KTOP 5            // k+1 = 5
#define MARGIN 0.1f
#define NEGV (-1.0e30f)

typedef __attribute__((ext_vector_type(2))) float v2f;
typedef __attribute__((ext_vector_type(8))) float v8f;

// ---------------------------------------------------------------------------
// Kernel 0: squared row norms.  One wave per row, lane loads float4, wave32
// xor-shuffle reduction.
// ---------------------------------------------------------------------------
__global__ void norms_kernel(const float* __restrict__ X, float* __restrict__ sq) {
    const int wave = threadIdx.x >> 5;
    const int lane = threadIdx.x & 31;
    const int row  = blockIdx.x * 4 + wave;
    const float4 v = reinterpret_cast<const float4*>(X + (size_t)row * D)[lane];
    float s = v.x * v.x + v.y * v.y + v.z * v.z + v.w * v.w;
    s += __shfl_xor(s, 16, 32);
    s += __shfl_xor(s, 8, 32);
    s += __shfl_xor(s, 4, 32);
    s += __shfl_xor(s, 2, 32);
    s += __shfl_xor(s, 1, 32);
    if (lane == 0) sq[row] = s;
}

// ---------------------------------------------------------------------------
// Kernel 1: S[i][j] = simi_hat = -dist(i,j) + (label_i != label_j)*MARGIN
// Gram matrix via V_WMMA_F32_16X16X4_F32 with 2x2 register blocking:
// each wave computes a 32x32 output region as four 16x16 WMMA tiles, so the
// inner k-loop issues 4 float2 loads feeding 4 WMMAs (1:1 load:WMMA ratio,
// operands reused across two tiles each).
//
// A (16x4 f32) lane layout: m = lane&15, holds K = 2*(lane>>4), 2*(lane>>4)+1.
// B (4x16 f32) mirrors with n = lane&15 -> for B = X^T both operands are
// loaded identically as row-major rows of X (aligned float2 loads).
// C/D (16x16 f32): n = lane&15, m = v + 8*(lane>>4) for VGPR v.
// ---------------------------------------------------------------------------
__global__ void simi_kernel(const float* __restrict__ X,
                            const int* __restrict__ labels,
                            const float* __restrict__ sq,
                            float* __restrict__ S) {
    const int lane = threadIdx.x & 31;
    const int wave = threadIdx.x >> 5;
    const int i0   = blockIdx.x * 32;
    const int j0   = (blockIdx.y * 8 + wave) * 32;
    const int mn   = lane & 15;          // A row / B col held by this lane
    const int kb   = (lane >> 4) * 2;    // K sub-offset: 0 or 2

    const float* a0 = X + (size_t)(i0 + mn) * D + kb;
    const float* a1 = X + (size_t)(i0 + 16 + mn) * D + kb;
    const float* b0 = X + (size_t)(j0 + mn) * D + kb;
    const float* b1 = X + (size_t)(j0 + 16 + mn) * D + kb;

    v8f acc00 = {}, acc01 = {}, acc10 = {}, acc11 = {};
    for (int k = 0; k < D; k += 4) {
        const v2f A0 = *reinterpret_cast<const v2f*>(a0 + k);
        const v2f A1 = *reinterpret_cast<const v2f*>(a1 + k);
        const v2f B0 = *reinterpret_cast<const v2f*>(b0 + k);
        const v2f B1 = *reinterpret_cast<const v2f*>(b1 + k);
        acc00 = __builtin_amdgcn_wmma_f32_16x16x4_f32(false, A0, false, B0,
                                                      (short)0, acc00, false, false);
        acc01 = __builtin_amdgcn_wmma_f32_16x16x4_f32(false, A0, false, B1,
                                                      (short)0, acc01, false, false);
        acc10 = __builtin_amdgcn_wmma_f32_16x16x4_f32(false, A1, false, B0,
                                                      (short)0, acc10, false, false);
        acc11 = __builtin_amdgcn_wmma_f32_16x16x4_f32(false, A1, false, B1,
                                                      (short)0, acc11, false, false);
    }

    const int mhi = 8 * (lane >> 4);

    // Epilogue: convert Gram products to margin-shifted similarities.
    #pragma unroll
    for (int tj = 0; tj < 2; ++tj) {
        const int   j   = j0 + 16 * tj + mn;
        const float sqj = sq[j];
        const int   lj  = labels[j];
        #pragma unroll
        for (int ti = 0; ti < 2; ++ti) {
            const v8f& acc = (tj == 0) ? (ti == 0 ? acc00 : acc10)
                                       : (ti == 0 ? acc01 : acc11);
            #pragma unroll
            for (int v = 0; v < 8; ++v) {
                const int   i    = i0 + 16 * ti + v + mhi;
                const float prod = acc[v];
                float d2 = fmaxf(sq[i] + sqj - 2.0f * prod, 0.0f);
                const float zero = (d2 == 0.0f) ? 1.0f : 0.0f;
                const float d    = sqrtf(d2 + zero * 1e-16f) * (1.0f - zero);
                const float nm   = (labels[i] == lj) ? 0.0f : 1.0f;  // 1 - cls_match
                S[(size_t)i * N + j] = -d + nm * MARGIN;
            }
        }
    }
}

// ---------------------------------------------------------------------------
// Kernel 2: per-row selection.  Thread per row r; reads column r of S
// (== row r by symmetry) so consecutive threads hit consecutive addresses.
// Maintains top-5 (value+flag) of the row and top-10 same-class values in
// registers; exact replacement for the reference's argsort/rank logic.
// ---------------------------------------------------------------------------
__global__ void select_kernel(const float* __restrict__ S,
                              const int* __restrict__ labels,
                              float* __restrict__ rowres) {
    const int r   = blockIdx.x * blockDim.x + threadIdx.x;
    const int myl = labels[r];

    float t5v[KTOP];
    int   t5f[KTOP];
    float ts[2 * KTOP];
    #pragma unroll
    for (int p = 0; p < KTOP; ++p) { t5v[p] = NEGV; t5f[p] = 0; }
    #pragma unroll
    for (int p = 0; p < 2 * KTOP; ++p) ts[p] = NEGV;

    int pos = 0;
    for (int j = 0; j < N; ++j) {
        const float val   = S[(size_t)j * N + r];
        const bool  match = (labels[j] == myl);
        pos += match ? 1 : 0;

        if (val > t5v[KTOP - 1]) {               // strict '>' => stable ties
            float v = val;
            int   f = match ? 1 : 0;
            #pragma unroll
            for (int p = 0; p < KTOP; ++p) {
                const bool  gt = v > t5v[p];
                const float hv = gt ? v : t5v[p];
                const int   hf = gt ? f : t5f[p];
                const float lv = gt ? t5v[p] : v;
                const int   lf = gt ? t5f[p] : f;
                t5v[p] = hv; t5f[p] = hf; v = lv; f = lf;
            }
        }
        if (match && val > ts[2 * KTOP - 1]) {
            float v = val;
            #pragma unroll
            for (int p = 0; p < 2 * KTOP; ++p) {
                const float hv = fmaxf(ts[p], v);
                const float lv = fminf(ts[p], v);
                ts[p] = hv; v = lv;
            }
        }
    }

    const int ks = pos < KTOP ? pos : KTOP;
    int   m = 0, fpn = 0;
    float fps = 0.0f;
    #pragma unroll
    for (int p = 0; p < KTOP; ++p) {
        if (p < ks) {
            if (t5f[p]) m++;
            else { fpn++; fps += t5v[p]; }
        }
    }
    // fn values sorted desc are ts[m], ts[m+1], ...  (top-m same-class are in
    // the global top-ks); sum the next fpn of them.
    float fns = 0.0f;
    #pragma unroll
    for (int q = 0; q < 2 * KTOP; ++q) {
        if (q >= m && q < m + fpn) fns += ts[q];
    }
    rowres[r] = fps - fns;
}

// ---------------------------------------------------------------------------
// Kernel 3: deterministic reduction of 4096 row results to one scalar.
// ---------------------------------------------------------------------------
__global__ void reduce_kernel(const float* __restrict__ rowres,
                              float* __restrict__ out) {
    __shared__ float sm[256];
    float s = 0.0f;
    for (int i = threadIdx.x; i < N; i += 256) s += rowres[i];
    sm[threadIdx.x] = s;
    __syncthreads();
    for (int off = 128; off > 0; off >>= 1) {
        if ((int)threadIdx.x < off) sm[threadIdx.x] += sm[threadIdx.x + off];
        __syncthreads();
    }
    if (threadIdx.x == 0) out[0] = sm[0];
}

// ---------------------------------------------------------------------------
extern "C" void kernel_launch(void* const* d_in, const int* in_sizes, int n_in,
                              void* d_out, int out_size, void* d_ws, size_t ws_size,
                              hipStream_t stream) {
    const float* X      = (const float*)d_in[0];
    const int*   labels = (const int*)d_in[1];

    char*  ws     = (char*)d_ws;
    float* sq     = (float*)ws;                    // 4096 f32
    float* rowres = (float*)(ws + 16 * 1024);      // 4096 f32
    float* S      = (float*)(ws + 32 * 1024);      // 4096*4096 f32 = 64 MB

    norms_kernel<<<N / 4, 128, 0, stream>>>(X, sq);

    dim3 g1(N / 32, N / (32 * 8));                 // 128 x 16 blocks, 8 waves each
    simi_kernel<<<g1, 256, 0, stream>>>(X, labels, sq, S);

    select_kernel<<<N / 256, 256, 0, stream>>>(S, labels, rowres);

    reduce_kernel<<<1, 256, 0, stream>>>(rowres, d_out ? (float*)d_out : (float*)nullptr);
}